// AffineTransformation_52158082842913
// MI455X (gfx1250) — compile-verified
//
#include <hip/hip_runtime.h>

typedef __attribute__((ext_vector_type(2))) float v2f;
typedef __attribute__((ext_vector_type(8))) float v8f;

#define BATCH_N 32
#define IMG_H 512
#define IMG_W 512
#define TILE 16
#define TILES_X (IMG_W / TILE)                 // 32
#define TILES_PER_IMG (TILES_X * (IMG_H/TILE)) // 1024
#define WAVES_PER_BLOCK 8
#define TOTAL_TILES (BATCH_N * TILES_PER_IMG)  // 32768

__global__ __launch_bounds__(256)
void AffineTransformation_gfx1250_kernel(const float* __restrict__ theta,
                                         const float* __restrict__ image,
                                         float* __restrict__ out)
{
    const int lane = threadIdx.x & 31;
    const int wave = threadIdx.x >> 5;

    // one 16x16 output tile per wave
    const int tile = blockIdx.x * WAVES_PER_BLOCK + wave;
    const int b    = tile / TILES_PER_IMG;
    const int rem  = tile - b * TILES_PER_IMG;
    const int i0   = (rem / TILES_X) * TILE;   // tile row origin
    const int j0   = (rem % TILES_X) * TILE;   // tile col origin

    const float t0 = theta[b * 6 + 0];
    const float t1 = theta[b * 6 + 1];
    const float t2 = theta[b * 6 + 2];
    const float t3 = theta[b * 6 + 3];
    const float t4 = theta[b * 6 + 4];
    const float t5 = theta[b * 6 + 5];

    const float sw   = (float)(IMG_W - 1) * 0.5f;   // 255.5
    const float sh   = (float)(IMG_H - 1) * 0.5f;   // 255.5
    const float stepx = 2.0f / (float)(IMG_W - 1);  // constant: no runtime divide
    const float stepy = 2.0f / (float)(IMG_H - 1);

    const int n  = lane & 15;          // column within tile (N index)
    const int hi = (lane >> 4) & 1;    // lane-half selects M base (0 or 8)

    // Column-dependent part of the rescaled source coords:
    //   srs_x(i,j) = sw*(t0*x_j) + sw*(t1*y_i + t2 + 1)
    //   srs_y(i,j) = sh*(t3*x_j) + sh*(t4*y_i + t5 + 1)
    const float xj   = fmaf((float)(j0 + n), stepx, -1.0f);
    const float colx = sw * (t0 * xj);
    const float coly = sh * (t3 * xj);

    // A = all ones (layout independent); B = one full 16-col row of column
    // terms striped over lanes 0-15 of VGPR0, zeros elsewhere. With A==1 the
    // K-dot picks up that row regardless of which K slot it maps to; the
    // remaining zero terms are exact. C carries the row-dependent part in the
    // C/D layout (VGPR v, lane L -> M = v + 8*(L>=16)).
    v2f amat; amat.x = 1.0f; amat.y = 1.0f;
    v2f bxv;  bxv.x = (lane < 16) ? colx : 0.0f; bxv.y = 0.0f;
    v2f byv;  byv.x = (lane < 16) ? coly : 0.0f; byv.y = 0.0f;

    v8f cxv, cyv;
#pragma unroll
    for (int v = 0; v < 8; ++v) {
        const int   m  = v + hi * 8;
        const float yi = fmaf((float)(i0 + m), stepy, -1.0f);
        cxv[v] = sw * (t1 * yi + t2 + 1.0f);
        cyv[v] = sh * (t4 * yi + t5 + 1.0f);
    }

    // D[m][n] = colpart(j0+n) + rowpart(i0+m)  == rescaled source coordinate
    v8f dx = __builtin_amdgcn_wmma_f32_16x16x4_f32(
        false, amat, false, bxv, (short)0, cxv, false, false);
    v8f dy = __builtin_amdgcn_wmma_f32_16x16x4_f32(
        false, amat, false, byv, (short)0, cyv, false, false);

    const float* __restrict__ img = image + (size_t)b * IMG_H * IMG_W;
    float*       __restrict__ dst = out   + (size_t)b * IMG_H * IMG_W;

#pragma unroll
    for (int v = 0; v < 8; ++v) {
        const int m = v + hi * 8;
        const int i = i0 + m;
        const int j = j0 + n;

        const float sx = dx[v];
        const float sy = dy[v];
        const float fx = floorf(sx);
        const float fy = floorf(sy);

        // Clamped 2x2 corner coordinates (shared by the 4 neighbors);
        // clamp happens BEFORE the weights, matching the reference.
        const float nx0 = fminf(fmaxf(fx,        0.0f), (float)(IMG_W - 1));
        const float nx1 = fminf(fmaxf(fx + 1.0f, 0.0f), (float)(IMG_W - 1));
        const float ny0 = fminf(fmaxf(fy,        0.0f), (float)(IMG_H - 1));
        const float ny1 = fminf(fmaxf(fy + 1.0f, 0.0f), (float)(IMG_H - 1));

        const float wx0 = fmaxf(0.0f, 1.0f - fabsf(sx - nx0));
        const float wx1 = fmaxf(0.0f, 1.0f - fabsf(sx - nx1));
        const float wy0 = fmaxf(0.0f, 1.0f - fabsf(sy - ny0));
        const float wy1 = fmaxf(0.0f, 1.0f - fabsf(sy - ny1));

        const int ix0 = (int)nx0;
        const int ix1 = (int)nx1;
        const int iy0 = (int)ny0;
        const int iy1 = (int)ny1;

        const int r0 = iy0 * IMG_W;
        const int r1 = iy1 * IMG_W;

        // Four independent gathers -> scheduler can issue all before waiting.
        const float p00 = img[r0 + ix0];
        const float p01 = img[r0 + ix1];
        const float p10 = img[r1 + ix0];
        const float p11 = img[r1 + ix1];

        float acc;
        acc = p00 * (wx0 * wy0);
        acc = fmaf(p01, wx1 * wy0, acc);
        acc = fmaf(p10, wx0 * wy1, acc);
        acc = fmaf(p11, wx1 * wy1, acc);

        dst[i * IMG_W + j] = fminf(fmaxf(acc, 0.0f), 1.0f);
    }
}

extern "C" void kernel_launch(void* const* d_in, const int* in_sizes, int n_in,
                              void* d_out, int out_size, void* d_ws, size_t ws_size,
                              hipStream_t stream) {
    (void)in_sizes; (void)n_in; (void)d_ws; (void)ws_size; (void)out_size;
    const float* theta = (const float*)d_in[0];
    const float* image = (const float*)d_in[1];
    float*       outp  = (float*)d_out;

    const int blocks = TOTAL_TILES / WAVES_PER_BLOCK;  // 4096
    AffineTransformation_gfx1250_kernel<<<blocks, 256, 0, stream>>>(theta, image, outp);
}